// Clustering_71536975282249
// MI455X (gfx1250) — compile-verified
//
#include <hip/hip_runtime.h>
#include <math.h>

typedef float v2f __attribute__((ext_vector_type(2)));
typedef float v4f __attribute__((ext_vector_type(4)));
typedef float v8f __attribute__((ext_vector_type(8)));

#define BATCH   8
#define HEADS   8
#define LQ      512
#define LKK     512
#define DK      64
#define CNUM    5
#define DMODEL  (HEADS * DK)          // 512
#define KFLAT   (LKK * DMODEL)        // 262144
#define SCALE   0.125f                // 1/sqrt(64)
#define LOG2PI_ 1.8378770664093453f

// ---------------------------------------------------------------------------
// Kernel 0: y1[b][n] = b1[n]  (init accumulator with bias)
// ---------------------------------------------------------------------------
__global__ void init_y1(const float* __restrict__ b1, float* __restrict__ y1) {
    y1[blockIdx.x * 512 + threadIdx.x] = b1[threadIdx.x];
}

// ---------------------------------------------------------------------------
// Kernel 1: y1 += flat(K) @ W1.   M=8, K=262144, N=512.  HBM-bound on W1
// (512 MB @ 23.3 TB/s ~ 23 us, the dominant cost of the whole problem).
// W1 reads are fully coalesced across the 512 threads; flat(K) chunks are
// staged into LDS with the CDNA5 async global->LDS DMA path (no VGPR
// round-trip, tracked by ASYNCcnt), then broadcast to all lanes.
// ---------------------------------------------------------------------------
#define GV_CHUNK 1024
#define GV_TILE  256
__global__ __launch_bounds__(512) void gemv_w1(const float* __restrict__ Kf,
                                               const float* __restrict__ W1,
                                               float* __restrict__ y1) {
    __shared__ float sflat[BATCH * GV_TILE];   // 8 KB
    const int n   = threadIdx.x;               // output column 0..511
    const int kk0 = blockIdx.x * GV_CHUNK;

    float acc[BATCH];
#pragma unroll
    for (int bb = 0; bb < BATCH; ++bb) acc[bb] = 0.0f;

    for (int t = 0; t < GV_CHUNK / GV_TILE; ++t) {
        const int kbase = kk0 + t * GV_TILE;
        // async DMA: global -> LDS, one dword per lane, 4 issues per wave
#pragma unroll
        for (int i = threadIdx.x; i < BATCH * GV_TILE; i += 512) {
            const int bb = i >> 8;             // /256
            const int kl = i & (GV_TILE - 1);
            const float* src = &Kf[(size_t)bb * KFLAT + kbase + kl];
            // LDS byte offset = low 32 bits of the flat address of __shared__
            const unsigned lds_off = (unsigned)(uintptr_t)&sflat[i];
            asm volatile("global_load_async_to_lds_b32 %0, %1, off"
                         :: "v"(lds_off), "v"(src)
                         : "memory");
        }
        asm volatile("s_wait_asynccnt 0x0" ::: "memory");
        __syncthreads();
        for (int kl = 0; kl < GV_TILE; ++kl) {
            const float wv = W1[(size_t)(kbase + kl) * 512 + n];
#pragma unroll
            for (int bb = 0; bb < BATCH; ++bb)
                acc[bb] += sflat[bb * GV_TILE + kl] * wv;
        }
        __syncthreads();
    }
#pragma unroll
    for (int bb = 0; bb < BATCH; ++bb)
        atomicAdd(&y1[bb * 512 + n], acc[bb]);
}

// ---------------------------------------------------------------------------
// Kernel 2: tiny head — x = relu(y1@W2+b2); cluster_q/k = softmax(x@W+b);
// loss (normal log-lik + soft-target CE); inds = argmax(cluster_q).
// One block, 64 threads; all tensors are 8x5.
// ---------------------------------------------------------------------------
__global__ void head_small(const float* __restrict__ y1,
                           const float* __restrict__ W2, const float* __restrict__ b2,
                           const float* __restrict__ Wq, const float* __restrict__ bq,
                           const float* __restrict__ Wkp, const float* __restrict__ bk,
                           float* __restrict__ loss_out, int* __restrict__ inds_out) {
    __shared__ float xs[BATCH][CNUM], lqs[BATCH][CNUM], lks[BATCH][CNUM];
    __shared__ float cqs[BATCH][CNUM], cks[BATCH][CNUM];
    const int tid = threadIdx.x;

    if (tid < BATCH * CNUM) {
        const int bb = tid / CNUM, c = tid % CNUM;
        float a = b2[c];
        for (int nn = 0; nn < 512; ++nn) a += y1[bb * 512 + nn] * W2[nn * CNUM + c];
        xs[bb][c] = fmaxf(a, 0.0f);
    }
    __syncthreads();
    if (tid < BATCH * CNUM) {
        const int bb = tid / CNUM, c = tid % CNUM;
        float aq = bq[c], ak = bk[c];
#pragma unroll
        for (int j = 0; j < CNUM; ++j) {
            aq += xs[bb][j] * Wq[j * CNUM + c];
            ak += xs[bb][j] * Wkp[j * CNUM + c];
        }
        lqs[bb][c] = aq; lks[bb][c] = ak;
    }
    __syncthreads();
    if (tid < BATCH) {
        const int bb = tid;
        float mq = -INFINITY, mk = -INFINITY;
#pragma unroll
        for (int c = 0; c < CNUM; ++c) { mq = fmaxf(mq, lqs[bb][c]); mk = fmaxf(mk, lks[bb][c]); }
        float sq = 0.0f, sk = 0.0f;
#pragma unroll
        for (int c = 0; c < CNUM; ++c) {
            cqs[bb][c] = __expf(lqs[bb][c] - mq); sq += cqs[bb][c];
            cks[bb][c] = __expf(lks[bb][c] - mk); sk += cks[bb][c];
        }
#pragma unroll
        for (int c = 0; c < CNUM; ++c) { cqs[bb][c] /= sq; cks[bb][c] /= sk; }
        int best = 0; float bv = cqs[bb][0];
#pragma unroll
        for (int c = 1; c < CNUM; ++c)
            if (cqs[bb][c] > bv) { bv = cqs[bb][c]; best = c; }
        inds_out[bb] = best;
    }
    __syncthreads();
    if (tid == 0) {
        float mu[CNUM], sigma[CNUM];
#pragma unroll
        for (int c = 0; c < CNUM; ++c) {
            float sq = 0.0f, skm = 0.0f;
            for (int b = 0; b < BATCH; ++b) { sq += cqs[b][c]; skm += cks[b][c]; }
            mu[c] = sq / 8.0f;
            const float mk = skm / 8.0f;
            float var = 0.0f;
            for (int b = 0; b < BATCH; ++b) { const float d = cks[b][c] - mk; var += d * d; }
            var /= 7.0f;                              // unbiased (ddof=1)
            const float sd = sqrtf(var);
            sigma[c] = log1pf(__expf(sd));            // softplus
        }
        float lp = 0.0f;
        for (int b = 0; b < BATCH; ++b)
#pragma unroll
            for (int c = 0; c < CNUM; ++c) {
                const float z = (cks[b][c] - mu[c]) / sigma[c];
                lp += -0.5f * z * z - logf(sigma[c]) - 0.5f * LOG2PI_;
            }
        const float loss1 = -lp / (float)(BATCH * CNUM);
        float ce = 0.0f;
        for (int b = 0; b < BATCH; ++b) {
            float mm = -INFINITY;
#pragma unroll
            for (int c = 0; c < CNUM; ++c) mm = fmaxf(mm, cqs[b][c]);
            float s = 0.0f;
#pragma unroll
            for (int c = 0; c < CNUM; ++c) s += __expf(cqs[b][c] - mm);
            const float lse = mm + logf(s);
            float acc = 0.0f;
#pragma unroll
            for (int c = 0; c < CNUM; ++c) acc += cqs[b][c] * (cqs[b][c] - lse);
            ce += -acc;
        }
        ce /= (float)BATCH;
        loss_out[0] = loss1 + ce;
    }
}

// ---------------------------------------------------------------------------
// Kernel 3: S = scale * Q K^T per (b,h) via V_WMMA_F32_16X16X4_F32.
// One wave per 16-row q-tile; Q tile (pre-scaled) held in 32 VGPRs across
// the whole k sweep.  A layout (16x4 f32): lane L -> m=L&15, holds d =
// 4*it + 2*(L>>4) + {0,1}.  B (4x16): lane L -> n=L&15, same d pair.
// D: n=L&15, row = r + 8*(L>>4).  S -> d_out (L2-resident for pass 2).
// ---------------------------------------------------------------------------
__global__ __launch_bounds__(256) void scores_wmma(const float* __restrict__ Q,
                                                   const float* __restrict__ K,
                                                   float* __restrict__ S) {
    const int wid  = (blockIdx.x * blockDim.x + threadIdx.x) >> 5; // 0..2047
    const int lane = threadIdx.x & 31;
    const int bh   = wid >> 5;        // 0..63
    const int qt   = wid & 31;        // q tile 0..31
    const int mn   = lane & 15;       // A-row / B-col / D-col index
    const int hi   = lane >> 4;

    const float* qrow = Q + ((size_t)bh * LQ + qt * 16 + mn) * DK + 2 * hi;
    v2f aq[16];
#pragma unroll
    for (int it = 0; it < 16; ++it) {
        aq[it] = *(const v2f*)(qrow + 4 * it);
        aq[it][0] *= SCALE;
        aq[it][1] *= SCALE;
    }

    for (int kt = 0; kt < 32; ++kt) {
        const float* krow = K + ((size_t)bh * LKK + kt * 16 + mn) * DK + 2 * hi;
        v8f acc = {0.f, 0.f, 0.f, 0.f, 0.f, 0.f, 0.f, 0.f};
#pragma unroll
        for (int it = 0; it < 16; ++it) {
            const v2f bm = *(const v2f*)(krow + 4 * it);
            acc = __builtin_amdgcn_wmma_f32_16x16x4_f32(
                /*neg_a=*/false, aq[it], /*neg_b=*/false, bm,
                /*c_mod=*/(short)0, acc, /*reuse_a=*/false, /*reuse_b=*/false);
        }
        float* srow = S + ((size_t)bh * LQ + qt * 16) * LKK + kt * 16 + mn;
#pragma unroll
        for (int r = 0; r < 8; ++r)
            srow[(size_t)(r + 8 * hi) * LKK] = acc[r];
    }
}

// ---------------------------------------------------------------------------
// Kernel 4: per row (b,h,q): P = softmax(tril-masked S); U = uniform over
// k>q (the all-zero cluster, present iff inds[b]>=1); final = max(S,P,[U]).
// One wave32 per row, in-place on S.  Lane-contiguous float4 mapping
// (k = j*128 + lane*4 + e) -> every load/store is a dense 512 B b128 wave
// transaction over the 67 MB (L2-resident) row buffer.
// ---------------------------------------------------------------------------
__global__ __launch_bounds__(256) void softmax_final(float* __restrict__ S,
                                                     const int* __restrict__ inds) {
    const int row  = (blockIdx.x * blockDim.x + threadIdx.x) >> 5; // 0..32767
    const int lane = threadIdx.x & 31;
    const int q    = row & (LQ - 1);
    const int b    = row >> 12;        // / (HEADS*LQ)
    float* base    = S + (size_t)row * LKK;

    v4f vv[4];
#pragma unroll
    for (int j = 0; j < 4; ++j)
        vv[j] = *(const v4f*)(base + j * 128 + lane * 4);

    float mloc = -INFINITY;
#pragma unroll
    for (int j = 0; j < 4; ++j)
#pragma unroll
        for (int e = 0; e < 4; ++e) {
            const int k = j * 128 + lane * 4 + e;
            mloc = fmaxf(mloc, (k <= q) ? -1e9f : vv[j][e]);
        }
#pragma unroll
    for (int off = 16; off > 0; off >>= 1)
        mloc = fmaxf(mloc, __shfl_xor(mloc, off, 32));

    float sloc = 0.0f;
#pragma unroll
    for (int j = 0; j < 4; ++j)
#pragma unroll
        for (int e = 0; e < 4; ++e) {
            const int k = j * 128 + lane * 4 + e;
            sloc += __expf(((k <= q) ? -1e9f : vv[j][e]) - mloc);
        }
#pragma unroll
    for (int off = 16; off > 0; off >>= 1)
        sloc += __shfl_xor(sloc, off, 32);
    const float inv = 1.0f / sloc;

    const bool  hasU  = inds[b] >= 1;
    const float ufull = (q == LQ - 1) ? (1.0f / (float)LQ)
                                      : (1.0f / (float)(LQ - 1 - q));
#pragma unroll
    for (int j = 0; j < 4; ++j) {
        v4f fo;
#pragma unroll
        for (int e = 0; e < 4; ++e) {
            const int k  = j * 128 + lane * 4 + e;
            const float p = __expf(((k <= q) ? -1e9f : vv[j][e]) - mloc) * inv;
            float f = fmaxf(vv[j][e], p);
            if (hasU) {
                const float u = (q == LQ - 1) ? ufull : ((k > q) ? ufull : 0.0f);
                f = fmaxf(f, u);
            }
            fo[e] = f;
        }
        *(v4f*)(base + j * 128 + lane * 4) = fo;
    }
}

// ---------------------------------------------------------------------------
extern "C" void kernel_launch(void* const* d_in, const int* in_sizes, int n_in,
                              void* d_out, int out_size, void* d_ws, size_t ws_size,
                              hipStream_t stream) {
    const float* Q   = (const float*)d_in[0];
    const float* K   = (const float*)d_in[1];
    const float* W1  = (const float*)d_in[2];
    const float* b1  = (const float*)d_in[3];
    const float* W2  = (const float*)d_in[4];
    const float* b2  = (const float*)d_in[5];
    const float* Wq  = (const float*)d_in[6];
    const float* bq  = (const float*)d_in[7];
    const float* Wkp = (const float*)d_in[8];
    const float* bk  = (const float*)d_in[9];

    float* out  = (float*)d_out;                         // [final (16.78M) | loss]
    float* loss = out + (size_t)out_size - 1;

    float* ws_y1   = (float*)d_ws;                       // 8*512 f32 = 16 KB
    int*   ws_inds = (int*)((char*)d_ws + BATCH * 512 * sizeof(float));

    // 1) y1 = b1 (broadcast init)
    init_y1<<<dim3(BATCH), dim3(512), 0, stream>>>(b1, ws_y1);
    // 2) y1 += flat(K) @ W1   (HBM-bound: 512 MB of W1 at 23.3 TB/s)
    gemv_w1<<<dim3(KFLAT / GV_CHUNK), dim3(512), 0, stream>>>(K, W1, ws_y1);
    // 3) tiny head: x, cluster_q/k, loss, argmax inds
    head_small<<<dim3(1), dim3(64), 0, stream>>>(ws_y1, W2, b2, Wq, bq, Wkp, bk,
                                                 loss, ws_inds);
    // 4) S = scale * Q K^T (fp32 WMMA), written into d_out (L2-resident)
    scores_wmma<<<dim3(256), dim3(256), 0, stream>>>(Q, K, out);
    // 5) in-place masked softmax + max-combine with uniform cluster
    softmax_final<<<dim3((BATCH * HEADS * LQ) / 8), dim3(256), 0, stream>>>(out, ws_inds);
}